// GNN_47502338294214
// MI455X (gfx1250) — compile-verified
//
#include <hip/hip_runtime.h>
#include <math.h>

// Problem constants (match reference)
#define NN 50000
#define FD 128
#define EE 800000
#define GG 512
#define CC 10

typedef float v2f __attribute__((ext_vector_type(2)));
typedef float v8f __attribute__((ext_vector_type(8)));

__device__ __forceinline__ v8f wmma_f32(v2f a, v2f b, v8f c) {
    // D = A(16x4,f32) * B(4x16,f32) + C(16x16,f32)
    return __builtin_amdgcn_wmma_f32_16x16x4_f32(
        false, a, false, b, (short)0, c, false, false);
}

// gfx1250 async copy: global -> LDS, 16 bytes per lane, tracked by ASYNCcnt.
// lds_off is the low-32-bit LDS offset (generic shared pointer truncated);
// per-lane LDS address comes from a VGPR so padded layouts are fine.
__device__ __forceinline__ void async_copy_b128(unsigned lds_off, const void* gp) {
    asm volatile("global_load_async_to_lds_b128 %0, %1, off"
                 :: "v"(lds_off), "v"((unsigned long long)(size_t)gp)
                 : "memory");
}
__device__ __forceinline__ void wait_async_zero() {
    asm volatile("s_wait_asynccnt 0" ::: "memory");
}

// ---------------------------------------------------------------------------
// Kernel: zero-fill scratch (float4 stores; callers guarantee n % 4 == 0)
// ---------------------------------------------------------------------------
__global__ void k_zero(float4* __restrict__ p, size_t n4) {
    size_t i = (size_t)blockIdx.x * blockDim.x + threadIdx.x;
    size_t stride = (size_t)gridDim.x * blockDim.x;
    const float4 z = {0.f, 0.f, 0.f, 0.f};
    for (; i < n4; i += stride) p[i] = z;
}

// ---------------------------------------------------------------------------
// Kernel: combine edge-path weights (tiny, plain VALU).
//   Wc[r][c] = sum_k Wl[r*128+k] * We[k*128+c]   (r in 0..255, c in 0..127)
//   bc[c]    = sum_k bl[k]       * We[k*128+c] + be[c]
// ---------------------------------------------------------------------------
__global__ __launch_bounds__(128)
void k_combine_w(const float* __restrict__ Wl, const float* __restrict__ bl,
                 const float* __restrict__ We, const float* __restrict__ be,
                 float* __restrict__ Wc, float* __restrict__ bc) {
    int c = threadIdx.x;
    int r = blockIdx.x;
    if (r < 256) {
        float s = 0.f;
        #pragma unroll 4
        for (int k = 0; k < 128; ++k) s += Wl[r * 128 + k] * We[k * 128 + c];
        Wc[r * 128 + c] = s;
    } else {
        float s = be[c];
        #pragma unroll 4
        for (int k = 0; k < 128; ++k) s += bl[k] * We[k * 128 + c];
        bc[c] = s;
    }
}

// ---------------------------------------------------------------------------
// Kernel: node GEMM producing ab[N][256]:
//   ab[n][0:128]   = h[n] @ Wc[0:128]   + post(h[n]) + bc     (a' part)
//   ab[n][128:256] = h[n] @ Wc[128:256]                        (b  part)
// One 512-thread block (16 waves) per 16-node strip; wave w -> column tile w.
// A strip staged once in LDS via async global->LDS copies (ASYNCcnt).
// ---------------------------------------------------------------------------
template<int RELU_POST>
__global__ __launch_bounds__(512)
void k_node_gemm(const float* __restrict__ h, const float* __restrict__ Wc,
                 const float* __restrict__ bc, float* __restrict__ ab) {
    __shared__ float as[16][132];        // padded stride: 132 floats (528 B)

    const int tid = threadIdx.x;
    const int r0 = blockIdx.x * 16;

    // Async-stage A strip (16 rows x 128 f32): each thread moves 16 bytes.
    {
        const int row = tid >> 5;            // 0..15
        const int col = (tid & 31) * 4;      // 0..124
        unsigned lds_off = (unsigned)(size_t)(&as[row][col]);
        async_copy_b128(lds_off, h + (size_t)(r0 + row) * FD + col);
        wait_async_zero();                   // this wave's copies complete
    }
    __syncthreads();                         // all waves' copies visible

    // Force wave id scalar so the epilogue branches are SCC-uniform.
    const int wave = __builtin_amdgcn_readfirstlane(tid >> 5);   // 0..15
    const int lane = tid & 31;
    const int lrow = lane & 15;
    const int half = lane >> 4;

    const bool is_a = (wave < 8);
    const float* Wb = Wc + (is_a ? 0 : (128 * 128));
    const int c0 = (wave & 7) * 16;

    const float* bp = Wb + (size_t)(2 * half) * 128 + c0 + lrow;

    v8f acc = {};
    #pragma unroll 8
    for (int k = 0; k < 128; k += 4) {
        v2f a; a.x = as[lrow][k + 2 * half]; a.y = as[lrow][k + 2 * half + 1];
        v2f b; b.x = bp[0];                  b.y = bp[128];
        acc = wmma_f32(a, b, acc);
        bp += 4 * 128;
    }

    const int colg = wave * 16 + lrow;
    if (is_a) {
        const float bias = bc[c0 + lrow];
        #pragma unroll
        for (int v = 0; v < 8; ++v) {
            int lr = v + 8 * half;           // local row 0..15
            float hp = as[lr][c0 + lrow];
            if (RELU_POST) hp = fmaxf(hp, 0.f);
            ab[(size_t)(r0 + lr) * 256 + colg] = acc[v] + hp + bias;
        }
    } else {
        #pragma unroll
        for (int v = 0; v < 8; ++v) {
            int lr = v + 8 * half;
            ab[(size_t)(r0 + lr) * 256 + colg] = acc[v];
        }
    }
}

// ---------------------------------------------------------------------------
// Kernel: edge phase. For each edge e: m = relu(a'[src] + b[dst]);
// atomically accumulate into aggr[dst]. One wave per edge, float4 per lane.
// ---------------------------------------------------------------------------
__global__ __launch_bounds__(256)
void k_edge(const int* __restrict__ src, const int* __restrict__ dst,
            const float* __restrict__ ab, float* __restrict__ aggr, int nE) {
    const int e = blockIdx.x * 8 + (threadIdx.x >> 5);
    if (e >= nE) return;
    const int lane = threadIdx.x & 31;
    const int s = src[e];
    const int d = dst[e];
    const float4 a = *(const float4*)(ab + (size_t)s * 256 + lane * 4);
    const float4 b = *(const float4*)(ab + (size_t)d * 256 + 128 + lane * 4);
    float4 m;
    m.x = fmaxf(a.x + b.x, 0.f);
    m.y = fmaxf(a.y + b.y, 0.f);
    m.z = fmaxf(a.z + b.z, 0.f);
    m.w = fmaxf(a.w + b.w, 0.f);
    float* o = aggr + (size_t)d * FD + lane * 4;
    __hip_atomic_fetch_add(o + 0, m.x, __ATOMIC_RELAXED, __HIP_MEMORY_SCOPE_AGENT);
    __hip_atomic_fetch_add(o + 1, m.y, __ATOMIC_RELAXED, __HIP_MEMORY_SCOPE_AGENT);
    __hip_atomic_fetch_add(o + 2, m.z, __ATOMIC_RELAXED, __HIP_MEMORY_SCOPE_AGENT);
    __hip_atomic_fetch_add(o + 3, m.w, __ATOMIC_RELAXED, __HIP_MEMORY_SCOPE_AGENT);
}

// ---------------------------------------------------------------------------
// Kernel: fused node MLP: z = post(h) + aggr; t = relu(z@W1 + b1);
// out = t@W2 + b2 (DO_POOL: relu + atomic global_add_pool instead of store).
// Block = 256 threads (8 waves) handles 16 nodes; z,t staged in padded LDS.
// ---------------------------------------------------------------------------
template<int RELU_IN, int DO_POOL>
__global__ __launch_bounds__(256)
void k_mlp(const float* __restrict__ h,
           const float* __restrict__ aggr,
           const float* __restrict__ W1, const float* __restrict__ b1,
           const float* __restrict__ W2, const float* __restrict__ b2,
           float* __restrict__ out,
           float* __restrict__ pooled, const int* __restrict__ batch) {
    __shared__ float zs[16][132];   // padded: stride 132 avoids bank conflicts
    __shared__ float ts[16][132];

    const int tid = threadIdx.x;
    const int r0 = blockIdx.x * 16;

    // Stage 0: build z in LDS (float4: 2 quads per thread; rows are 16B-aligned)
    #pragma unroll
    for (int j = tid; j < 16 * 32; j += 256) {
        int rr = j >> 5, cc = (j & 31) * 4;
        float4 hv = *(const float4*)(h    + (size_t)(r0 + rr) * FD + cc);
        float4 av = *(const float4*)(aggr + (size_t)(r0 + rr) * FD + cc);
        if (RELU_IN) {
            hv.x = fmaxf(hv.x, 0.f); hv.y = fmaxf(hv.y, 0.f);
            hv.z = fmaxf(hv.z, 0.f); hv.w = fmaxf(hv.w, 0.f);
        }
        float4 zv; zv.x = hv.x + av.x; zv.y = hv.y + av.y;
        zv.z = hv.z + av.z; zv.w = hv.w + av.w;
        *(float4*)(&zs[rr][cc]) = zv;
    }
    __syncthreads();

    const int wave = __builtin_amdgcn_readfirstlane(tid >> 5);
    const int lane = tid & 31;
    const int lrow = lane & 15;
    const int half = lane >> 4;
    const int c0 = wave * 16;

    // Stage 1: t = relu(z @ W1 + b1)
    {
        v8f acc = {};
        const float* bp = W1 + (size_t)(2 * half) * 128 + c0 + lrow;
        #pragma unroll 8
        for (int k = 0; k < 128; k += 4) {
            v2f a; a.x = zs[lrow][k + 2 * half]; a.y = zs[lrow][k + 2 * half + 1];
            v2f b; b.x = bp[0]; b.y = bp[128];
            acc = wmma_f32(a, b, acc);
            bp += 4 * 128;
        }
        const float bias = b1[c0 + lrow];
        #pragma unroll
        for (int v = 0; v < 8; ++v)
            ts[v + 8 * half][c0 + lrow] = fmaxf(acc[v] + bias, 0.f);
    }
    __syncthreads();

    // Stage 2: out = t @ W2 + b2
    {
        v8f acc = {};
        const float* bp = W2 + (size_t)(2 * half) * 128 + c0 + lrow;
        #pragma unroll 8
        for (int k = 0; k < 128; k += 4) {
            v2f a; a.x = ts[lrow][k + 2 * half]; a.y = ts[lrow][k + 2 * half + 1];
            v2f b; b.x = bp[0]; b.y = bp[128];
            acc = wmma_f32(a, b, acc);
            bp += 4 * 128;
        }
        const float bias = b2[c0 + lrow];
        if (!DO_POOL) {
            #pragma unroll
            for (int v = 0; v < 8; ++v) {
                int row = r0 + v + 8 * half;
                out[(size_t)row * FD + c0 + lrow] = acc[v] + bias;
            }
        } else {
            // x2 = relu(out); pooled[batch[n]] += x2[n]  (global_add_pool fused)
            #pragma unroll
            for (int v = 0; v < 8; ++v) {
                int row = r0 + v + 8 * half;
                float val = fmaxf(acc[v] + bias, 0.f);
                int g = batch[row];
                __hip_atomic_fetch_add(&pooled[(size_t)g * FD + c0 + lrow], val,
                                       __ATOMIC_RELAXED, __HIP_MEMORY_SCOPE_AGENT);
            }
        }
    }
}

// ---------------------------------------------------------------------------
// Kernel: readout. One wave per graph: logits = pooled[g] @ W_lin1 + b_lin1,
// then log_softmax over 10 classes via wave shuffles.
// ---------------------------------------------------------------------------
__global__ __launch_bounds__(256)
void k_readout(const float* __restrict__ pooled, const float* __restrict__ W,
               const float* __restrict__ b, float* __restrict__ out, int nG) {
    const int g = blockIdx.x * 8 + (threadIdx.x >> 5);
    if (g >= nG) return;
    const int lane = threadIdx.x & 31;

    float logit = 0.f;
    float m = -INFINITY;
    if (lane < CC) {
        float s = b[lane];
        const float* pr = pooled + (size_t)g * FD;
        #pragma unroll 4
        for (int k = 0; k < 128; ++k) s += pr[k] * W[k * CC + lane];
        logit = s;
        m = s;
    }
    #pragma unroll
    for (int o = 16; o > 0; o >>= 1) m = fmaxf(m, __shfl_xor(m, o, 32));
    float e = (lane < CC) ? expf(logit - m) : 0.f;
    float ssum = e;
    #pragma unroll
    for (int o = 16; o > 0; o >>= 1) ssum += __shfl_xor(ssum, o, 32);
    if (lane < CC) out[(size_t)g * CC + lane] = logit - m - logf(ssum);
}

// ---------------------------------------------------------------------------
// Host launcher
// ---------------------------------------------------------------------------
extern "C" void kernel_launch(void* const* d_in, const int* in_sizes, int n_in,
                              void* d_out, int out_size, void* d_ws, size_t ws_size,
                              hipStream_t stream) {
    (void)in_sizes; (void)n_in; (void)out_size; (void)ws_size;

    const float* x       = (const float*)d_in[0];
    const int*   eidx    = (const int*)d_in[1];
    const int*   batch   = (const int*)d_in[2];
    const float* W_lin2  = (const float*)d_in[3];
    const float* b_lin2  = (const float*)d_in[4];
    const float* W_lin3  = (const float*)d_in[5];
    const float* b_lin3  = (const float*)d_in[6];
    const float* W_lin1  = (const float*)d_in[7];
    const float* b_lin1  = (const float*)d_in[8];
    const float* We1     = (const float*)d_in[9];
    const float* be1     = (const float*)d_in[10];
    const float* W11     = (const float*)d_in[11];
    const float* b11     = (const float*)d_in[12];
    const float* W12     = (const float*)d_in[13];
    const float* b12     = (const float*)d_in[14];
    const float* We2     = (const float*)d_in[15];
    const float* be2     = (const float*)d_in[16];
    const float* W21     = (const float*)d_in[17];
    const float* b21     = (const float*)d_in[18];
    const float* W22     = (const float*)d_in[19];
    const float* b22     = (const float*)d_in[20];

    const int* src = eidx;           // edge_index[0]
    const int* dst = eidx + EE;      // edge_index[1]

    // Workspace layout (floats)
    float* ws     = (float*)d_ws;
    float* ab     = ws;                              // N*256
    float* aggr   = ab   + (size_t)NN * 256;         // N*128
    float* x1     = aggr + (size_t)NN * FD;          // N*128
    float* Wc1    = x1   + (size_t)NN * FD;          // 256*128
    float* bc1    = Wc1  + 256 * 128;                // 128
    float* Wc2    = bc1  + 128;                      // 256*128
    float* bc2    = Wc2  + 256 * 128;                // 128
    float* pooled = bc2  + 128;                      // G*128

    const dim3 blk256(256);
    const dim3 blk512(512);

    // 0) zero aggr + pooled
    k_zero<<<1024, blk256, 0, stream>>>((float4*)aggr, (size_t)NN * FD / 4);
    k_zero<<<64,   blk256, 0, stream>>>((float4*)pooled, (size_t)GG * FD / 4);

    // 1) fold edge-linear + edge-projection weights
    k_combine_w<<<257, 128, 0, stream>>>(W_lin2, b_lin2, We1, be1, Wc1, bc1);
    k_combine_w<<<257, 128, 0, stream>>>(W_lin3, b_lin3, We2, be2, Wc2, bc2);

    // ---- conv1 ----
    k_node_gemm<0><<<NN / 16, blk512, 0, stream>>>(x, Wc1, bc1, ab);
    k_edge<<<EE / 8, blk256, 0, stream>>>(src, dst, ab, aggr, EE);
    k_mlp<0, 0><<<NN / 16, blk256, 0, stream>>>(x, aggr, W11, b11, W12, b12,
                                                x1, nullptr, nullptr);

    // ---- conv2 ----
    k_zero<<<1024, blk256, 0, stream>>>((float4*)aggr, (size_t)NN * FD / 4);
    k_node_gemm<1><<<NN / 16, blk512, 0, stream>>>(x1, Wc2, bc2, ab);
    k_edge<<<EE / 8, blk256, 0, stream>>>(src, dst, ab, aggr, EE);
    k_mlp<1, 1><<<NN / 16, blk256, 0, stream>>>(x1, aggr, W21, b21, W22, b22,
                                                nullptr, pooled, batch);

    // ---- readout ----
    k_readout<<<GG / 8, blk256, 0, stream>>>(pooled, W_lin1, b_lin1, (float*)d_out, GG);
}